// Complex2LayerMAPGraphConvolution_13606456393911
// MI455X (gfx1250) — compile-verified
//
#include <hip/hip_runtime.h>
#include <math.h>

typedef __attribute__((ext_vector_type(2))) float v2f;
typedef __attribute__((ext_vector_type(8))) float v8f;

#define FEAT 256
#define HIDW 256
#define OUTW 64

// ---------------------------------------------------------------------------
// 1) per-edge complex weights: w_r = w*cos(q*(ent+ccf)), w_i = w*sin(...)
// ---------------------------------------------------------------------------
__global__ void edge_weights_kernel(const float* __restrict__ wsym,
                                    const float* __restrict__ ent,
                                    const float* __restrict__ ccf,
                                    const float* __restrict__ q,
                                    float* __restrict__ wr,
                                    float* __restrict__ wi,
                                    int E) {
  int e = blockIdx.x * blockDim.x + threadIdx.x;
  if (e >= E) return;
  float ph = q[0] * (ent[e] + ccf[e]);
  float s, c;
  sincosf(ph, &s, &c);
  float w = wsym[e];
  wr[e] = w * c;
  wi[e] = w * s;
}

// ---------------------------------------------------------------------------
// 2) zero the scatter accumulators
// ---------------------------------------------------------------------------
__global__ void zero_kernel(float* __restrict__ p, size_t n) {
  size_t i = (size_t)blockIdx.x * blockDim.x + threadIdx.x;
  size_t stride = (size_t)gridDim.x * blockDim.x;
  for (; i < n; i += stride) p[i] = 0.0f;
}

// ---------------------------------------------------------------------------
// 3) fused complex SPMM: one pass over edges produces BOTH
//    S_r = A_r X_r - A_i X_i   and   S_i = A_i X_r + A_r X_i
//    one wave per edge; lane handles 8 of the 256 feature floats (2x float4)
// ---------------------------------------------------------------------------
__global__ void spmm_complex_kernel(const int* __restrict__ rowIdx,
                                    const int* __restrict__ colIdx,
                                    const float* __restrict__ wr,
                                    const float* __restrict__ wi,
                                    const float* __restrict__ Xr,
                                    const float* __restrict__ Xi,
                                    float* __restrict__ Sr,
                                    float* __restrict__ Si,
                                    int E) {
  int wave = (int)(((size_t)blockIdx.x * blockDim.x + threadIdx.x) >> 5);
  int lane = threadIdx.x & 31;
  if (wave >= E) return;
  int r = rowIdx[wave];
  int c = colIdx[wave];
  float a = wr[wave];
  float b = wi[wave];
  const float4* xr = (const float4*)(Xr + (size_t)c * FEAT);
  const float4* xi = (const float4*)(Xi + (size_t)c * FEAT);
  float* sr = Sr + (size_t)r * FEAT;
  float* si = Si + (size_t)r * FEAT;
#pragma unroll
  for (int j = 0; j < FEAT / 128; ++j) {
    int f4 = lane + j * 32;
    float4 vr = xr[f4];
    float4 vi = xi[f4];
    int base = f4 * 4;
    atomicAdd(sr + base + 0, a * vr.x - b * vi.x);
    atomicAdd(sr + base + 1, a * vr.y - b * vi.y);
    atomicAdd(sr + base + 2, a * vr.z - b * vi.z);
    atomicAdd(sr + base + 3, a * vr.w - b * vi.w);
    atomicAdd(si + base + 0, b * vr.x + a * vi.x);
    atomicAdd(si + base + 1, b * vr.y + a * vi.y);
    atomicAdd(si + base + 2, b * vr.z + a * vi.z);
    atomicAdd(si + base + 3, b * vr.w + a * vi.w);
  }
}

// ---------------------------------------------------------------------------
// 4) fused layer GEMM + comrelu, f32 WMMA (16x16x4), LDS-staged weights.
//    Block = 8 waves: one N-tile (16 output cols), 8 consecutive M-tiles.
//    The 16x256 W slice is staged ONCE per block into LDS via
//    global_load_async_to_lds_b64, pre-swizzled into WMMA-B fragment order:
//      shB word [ks*64 + kg*32 + n*2 + c] = W[tn*16+n][4*ks + 2*kg + c]
//    so the per-K-step B fragment is one conflict-free ds_load_b64 at
//    byte offset ks*256 + lane*8.
//    Z_r = S_r W^T        (bias cancels:  rr-ii)
//    Z_i = S_i W^T + 2*b  (bias doubles:  ir+ri)
//    mask = Z_r >= 0 ;  H_r = Z_r*mask ; H_i = Z_i*mask
// ---------------------------------------------------------------------------
__global__ void gemm_relu_lds_kernel(const float* __restrict__ Sr,
                                     const float* __restrict__ Si,
                                     const float* __restrict__ W,     // [256,256]
                                     const float* __restrict__ bvec,  // [256]
                                     float* __restrict__ Hr,
                                     float* __restrict__ Hi,
                                     int nnodes) {
  __shared__ float shB[64 * 64];  // 64 K-steps x 64 words = 16 KB

  const int NT = HIDW / 16;  // 16 column tiles
  int mblk = blockIdx.x / NT;
  int nblk = blockIdx.x % NT;
  int wave = threadIdx.x >> 5;
  int lane = threadIdx.x & 31;

  // ---- async-stage swizzled W tile: 2048 float2 pairs, 8 per wave ----
  {
    const float* Wt = W + (size_t)(nblk * 16) * HIDW;
    unsigned base = (unsigned)(uintptr_t)&shB[0];  // LDS byte offset (addr[31:0])
#pragma unroll
    for (int it = 0; it < 8; ++it) {
      int p = (wave * 8 + it) * 32 + lane;  // pair index 0..2047
      int n = p >> 7;                       // W row within tile (0..15)
      int kp = p & 127;                     // float2 pair along K
      int ks = kp >> 1;
      int kg = kp & 1;
      unsigned dst = base + (unsigned)((ks * 64 + kg * 32 + n * 2) * 4);
      unsigned long long src =
          (unsigned long long)(uintptr_t)(Wt + (size_t)n * HIDW + kp * 2);
      asm volatile("global_load_async_to_lds_b64 %0, %1, off"
                   :
                   : "v"(dst), "v"(src)
                   : "memory");
    }
    asm volatile("s_wait_asynccnt 0x0" ::: "memory");
    __syncthreads();
  }

  int tile_m = mblk * 8 + wave;
  if (tile_m * 16 < nnodes) {  // wave-uniform: EXEC all-1 for WMMA
    int nn = lane & 15;        // A-row within tile / B-col within tile
    int kg2 = (lane >> 4) << 1;
    int arow = tile_m * 16 + nn;
    if (arow >= nnodes) arow = nnodes - 1;  // clamp loads; stores guarded

    const float* pr = Sr + (size_t)arow * HIDW + kg2;
    const float* pi = Si + (size_t)arow * HIDW + kg2;
    const float* pb = shB + lane * 2;  // fragment-major LDS stream

    v8f accr = {};
    v8f acci = {};
    for (int ks = 0; ks < 64; ++ks) {
      v2f ar = *(const v2f*)(pr + ks * 4);
      v2f ai = *(const v2f*)(pi + ks * 4);
      v2f bb = *(const v2f*)(pb + ks * 64);
      accr = __builtin_amdgcn_wmma_f32_16x16x4_f32(false, ar, false, bb,
                                                   (short)0, accr, false, false);
      acci = __builtin_amdgcn_wmma_f32_16x16x4_f32(false, ai, false, bb,
                                                   (short)0, acci, false, false);
    }

    int ng = nblk * 16 + nn;
    float bias_i = 2.0f * bvec[ng];
    int mbase = tile_m * 16 + ((lane >> 4) << 3);  // D VGPR v -> row mbase+v
#pragma unroll
    for (int v = 0; v < 8; ++v) {
      int m = mbase + v;
      if (m < nnodes) {
        float zr = accr[v];
        float zi = acci[v] + bias_i;
        float mk = (zr >= 0.0f) ? 1.0f : 0.0f;
        size_t idx = (size_t)m * HIDW + ng;
        Hr[idx] = zr * mk;
        Hi[idx] = zi * mk;
      }
    }
  }
}

// ---------------------------------------------------------------------------
// 5) head: out = [l2r | l2i] @ W3^T + b3     (N x 512) @ (512 x 64)
//    split K: K=0..255 from Hr with W3[:, :256], K=256..511 from Hi.
// ---------------------------------------------------------------------------
__global__ void head_gemm_kernel(const float* __restrict__ Hr,
                                 const float* __restrict__ Hi,
                                 const float* __restrict__ W3,  // [64,512]
                                 const float* __restrict__ b3,  // [64]
                                 float* __restrict__ Out,       // [N,64]
                                 int nnodes) {
  const int NT = OUTW / 16;  // 4 column tiles
  int wave = (int)(((size_t)blockIdx.x * blockDim.x + threadIdx.x) >> 5);
  int lane = threadIdx.x & 31;
  int tile_m = wave / NT;
  int tile_n = wave % NT;
  if (tile_m * 16 >= nnodes) return;

  int nn = lane & 15;
  int kg = (lane >> 4) << 1;
  int arow = tile_m * 16 + nn;
  if (arow >= nnodes) arow = nnodes - 1;

  const float* pr = Hr + (size_t)arow * HIDW + kg;
  const float* pi = Hi + (size_t)arow * HIDW + kg;
  const float* pw = W3 + (size_t)(tile_n * 16 + nn) * (2 * HIDW) + kg;

  v8f acc = {};
  for (int k = 0; k < HIDW; k += 4) {
    __builtin_prefetch(pw + k + 64, 0, 1);
    v2f a = *(const v2f*)(pr + k);
    v2f bb = *(const v2f*)(pw + k);
    acc = __builtin_amdgcn_wmma_f32_16x16x4_f32(false, a, false, bb,
                                                (short)0, acc, false, false);
  }
  for (int k = 0; k < HIDW; k += 4) {
    v2f a = *(const v2f*)(pi + k);
    v2f bb = *(const v2f*)(pw + HIDW + k);
    acc = __builtin_amdgcn_wmma_f32_16x16x4_f32(false, a, false, bb,
                                                (short)0, acc, false, false);
  }

  int ng = tile_n * 16 + nn;
  float bias = b3[ng];
  int mbase = tile_m * 16 + ((lane >> 4) << 3);
#pragma unroll
  for (int v = 0; v < 8; ++v) {
    int m = mbase + v;
    if (m < nnodes) Out[(size_t)m * OUTW + ng] = acc[v] + bias;
  }
}

// ---------------------------------------------------------------------------
extern "C" void kernel_launch(void* const* d_in, const int* in_sizes, int n_in,
                              void* d_out, int out_size, void* d_ws, size_t ws_size,
                              hipStream_t stream) {
  const float* Xr   = (const float*)d_in[0];
  const float* Xi   = (const float*)d_in[1];
  const int*   row  = (const int*)d_in[2];
  const int*   col  = (const int*)d_in[3];
  const float* wsym = (const float*)d_in[4];
  const float* ent  = (const float*)d_in[5];
  const float* ccf  = (const float*)d_in[6];
  const float* q    = (const float*)d_in[7];
  const float* W1   = (const float*)d_in[8];
  const float* b1   = (const float*)d_in[9];
  const float* W2   = (const float*)d_in[10];
  const float* b2   = (const float*)d_in[11];
  const float* W3   = (const float*)d_in[12];
  const float* b3   = (const float*)d_in[13];

  const int E = in_sizes[2];
  const int N = in_sizes[0] / FEAT;

  // workspace layout (floats): wr[E] wi[E] Sr[N*256] Si[N*256] Hr[N*256] Hi[N*256]
  float* ws = (float*)d_ws;
  float* wr = ws;
  float* wi = wr + E;
  float* Sr = wi + E;
  float* Si = Sr + (size_t)N * FEAT;
  float* Hr = Si + (size_t)N * FEAT;
  float* Hi = Hr + (size_t)N * FEAT;

  const size_t zn = (size_t)2 * N * FEAT;  // Sr,Si contiguous
  const int spmm_blocks = (E + 7) / 8;     // 8 waves (edges) per 256-thr block
  const int mblocks = (N + 127) / 128;     // 8 M-tiles per block
  const int layer_blocks = mblocks * (HIDW / 16);
  const int head_waves = ((N + 15) / 16) * (OUTW / 16);

  edge_weights_kernel<<<(E + 255) / 256, 256, 0, stream>>>(wsym, ent, ccf, q, wr, wi, E);

  // ---- layer 1 ----
  zero_kernel<<<2048, 256, 0, stream>>>(Sr, zn);
  spmm_complex_kernel<<<spmm_blocks, 256, 0, stream>>>(row, col, wr, wi, Xr, Xi, Sr, Si, E);
  gemm_relu_lds_kernel<<<layer_blocks, 256, 0, stream>>>(Sr, Si, W1, b1, Hr, Hi, N);

  // ---- layer 2 ----
  zero_kernel<<<2048, 256, 0, stream>>>(Sr, zn);
  spmm_complex_kernel<<<spmm_blocks, 256, 0, stream>>>(row, col, wr, wi, Hr, Hi, Sr, Si, E);
  gemm_relu_lds_kernel<<<layer_blocks, 256, 0, stream>>>(Sr, Si, W2, b2, Hr, Hi, N);

  // ---- head ----
  head_gemm_kernel<<<(head_waves + 7) / 8, 256, 0, stream>>>(Hr, Hi, W3, b3, (float*)d_out, N);
}